// GraphAttentionLayer_59871844106789
// MI455X (gfx1250) — compile-verified
//
#include <hip/hip_runtime.h>
#include <hip/hip_bf16.h>

typedef __attribute__((ext_vector_type(2))) float v2f;
typedef __attribute__((ext_vector_type(8))) float v8f;

#define IN_F 256
#define OUT_F 128
#define ALPHA 0.2f

// ---------- helpers ----------
__device__ __forceinline__ int float_to_ordered_int(float f) {
    int i = __float_as_int(f);
    return (i >= 0) ? i : (i ^ 0x7FFFFFFF);
}
__device__ __forceinline__ float ordered_int_to_float(int i) {
    return __int_as_float((i >= 0) ? i : (i ^ 0x7FFFFFFF));
}
__device__ __forceinline__ void atomicAddF(float* p, float v) {
    unsafeAtomicAdd(p, v);   // -> global_atomic_add_f32
}

// ---------- 1) Wh = h @ W via f32 WMMA (16x16x4) ----------
// One wave per 16-row tile of h; computes all 8 col-tiles (128 cols).
__global__ void gat_gemm_wmma(const float* __restrict__ h,
                              const float* __restrict__ W,
                              float* __restrict__ Wh,
                              int nRows, int nTiles) {
    const int wave = threadIdx.x >> 5;
    const int lane = threadIdx.x & 31;
    const int tileM = blockIdx.x * 8 + wave;
    if (tileM >= nTiles) return;                 // wave-uniform

    const int halfSel = lane >> 4;               // 0: lanes 0-15, 1: lanes 16-31
    const int l15 = lane & 15;

    // A fragment row (clamped so EXEC stays all-ones)
    int aRowIdx = tileM * 16 + l15;
    if (aRowIdx >= nRows) aRowIdx = nRows - 1;
    const float* aRow = h + (size_t)aRowIdx * IN_F + halfSel * 2;

    v8f c[8];
    #pragma unroll
    for (int t = 0; t < 8; ++t) c[t] = (v8f){0.f,0.f,0.f,0.f,0.f,0.f,0.f,0.f};

    for (int k0 = 0; k0 < IN_F; k0 += 4) {
        v2f a;
        a.x = aRow[k0 + 0];
        a.y = aRow[k0 + 1];
        const int kb = k0 + halfSel * 2;
        #pragma unroll
        for (int t = 0; t < 8; ++t) {
            const int n = t * 16 + l15;
            v2f b;
            b.x = W[(size_t)(kb + 0) * OUT_F + n];
            b.y = W[(size_t)(kb + 1) * OUT_F + n];
            c[t] = __builtin_amdgcn_wmma_f32_16x16x4_f32(
                false, a, false, b, (short)0, c[t], false, false);
        }
    }

    // Store: lane half selects M rows 0-7 vs 8-15 of the tile
    const int rowBase = tileM * 16 + halfSel * 8;
    #pragma unroll
    for (int t = 0; t < 8; ++t) {
        const int col = t * 16 + l15;
        #pragma unroll
        for (int r = 0; r < 8; ++r) {
            const int row = rowBase + r;
            if (row < nRows)
                Wh[(size_t)row * OUT_F + col] = c[t][r];
        }
    }
}

// ---------- 2) per-node scores s[i]=Wh[i].a_i, t[i]=Wh[i].a_j ----------
__global__ void gat_scores(const float* __restrict__ Wh,
                           const float* __restrict__ a,
                           float* __restrict__ s, float* __restrict__ t,
                           int n) {
    const int node = blockIdx.x * 8 + (threadIdx.x >> 5);
    const int lane = threadIdx.x & 31;
    if (node >= n) return;
    const float4 v  = ((const float4*)(Wh + (size_t)node * OUT_F))[lane];
    const float4 ai = ((const float4*)a)[lane];
    const float4 aj = ((const float4*)(a + OUT_F))[lane];
    float ps = v.x*ai.x + v.y*ai.y + v.z*ai.z + v.w*ai.w;
    float pt = v.x*aj.x + v.y*aj.y + v.z*aj.z + v.w*aj.w;
    #pragma unroll
    for (int off = 16; off > 0; off >>= 1) {
        ps += __shfl_xor(ps, off, 32);
        pt += __shfl_xor(pt, off, 32);
    }
    if (lane == 0) { s[node] = ps; t[node] = pt; }
}

// ---------- 3) init workspace ----------
__global__ void gat_init(int* __restrict__ seg_max, float* __restrict__ seg_sum,
                         float* __restrict__ acc, int n) {
    const int i = blockIdx.x * blockDim.x + threadIdx.x;
    const int total = n * OUT_F;
    if (i < total) acc[i] = 0.f;
    if (i < n) { seg_max[i] = (int)0x80000000; seg_sum[i] = 0.f; }
}

// ---------- 4) edge pass 1: leaky-relu score + segment max ----------
__global__ void gat_edge_max(const int* __restrict__ ei,
                             const float* __restrict__ s,
                             const float* __restrict__ t,
                             float* __restrict__ e_leaky,
                             int* __restrict__ seg_max, int nE) {
    const int e = blockIdx.x * blockDim.x + threadIdx.x;
    if (e >= nE) return;
    const int src = ei[e];
    const int dst = ei[nE + e];
    const float raw = s[src] + t[dst];
    const float lk = raw > 0.f ? raw : ALPHA * raw;
    e_leaky[e] = lk;
    atomicMax(&seg_max[src], float_to_ordered_int(lk));
}

// ---------- 5) edge pass 2: fused exp + weighted scatter-add ----------
// One wave per edge; lane handles 4 of 128 features.
__global__ void gat_edge_agg(const int* __restrict__ ei,
                             const float* __restrict__ e_leaky,
                             const int* __restrict__ seg_max,
                             const float* __restrict__ Wh,
                             float* __restrict__ seg_sum,
                             float* __restrict__ acc, int nE) {
    const int e = blockIdx.x * 8 + (threadIdx.x >> 5);
    const int lane = threadIdx.x & 31;
    if (e >= nE) return;
    const int src = ei[e];
    const int dst = ei[nE + e];
    const float m = ordered_int_to_float(seg_max[src]);
    const float w = __expf(e_leaky[e] - m);
    if (lane == 0) atomicAddF(&seg_sum[src], w);
    const float4 v = ((const float4*)(Wh + (size_t)dst * OUT_F))[lane];
    float* d = acc + (size_t)src * OUT_F + lane * 4;
    atomicAddF(d + 0, w * v.x);
    atomicAddF(d + 1, w * v.y);
    atomicAddF(d + 2, w * v.z);
    atomicAddF(d + 3, w * v.w);
}

// ---------- 6) finalize: normalize + ELU ----------
__global__ void gat_finalize(const float* __restrict__ acc,
                             const float* __restrict__ seg_sum,
                             float* __restrict__ out, int n) {
    const int i = blockIdx.x * blockDim.x + threadIdx.x;
    if (i >= n * OUT_F) return;
    const int node = i >> 7;  // /128
    const float ssum = seg_sum[node];
    const float v = (ssum > 0.f) ? (acc[i] / ssum) : 0.f;
    out[i] = (v > 0.f) ? v : (__expf(v) - 1.f);
}

extern "C" void kernel_launch(void* const* d_in, const int* in_sizes, int n_in,
                              void* d_out, int out_size, void* d_ws, size_t ws_size,
                              hipStream_t stream) {
    const float* h  = (const float*)d_in[0];
    const int*   ei = (const int*)d_in[1];
    const float* W  = (const float*)d_in[2];
    const float* a  = (const float*)d_in[3];
    float* out = (float*)d_out;

    const int N  = in_sizes[0] / IN_F;
    const int nE = in_sizes[1] / 2;

    // workspace layout (all offsets multiples of 16 bytes)
    char* ws = (char*)d_ws;
    float* Wh      = (float*)ws;                  ws += (size_t)N * OUT_F * 4;
    float* acc     = (float*)ws;                  ws += (size_t)N * OUT_F * 4;
    float* s       = (float*)ws;                  ws += (size_t)N * 4;
    float* t       = (float*)ws;                  ws += (size_t)N * 4;
    float* seg_sum = (float*)ws;                  ws += (size_t)N * 4;
    int*   seg_max = (int*)ws;                    ws += (size_t)N * 4;
    float* e_leaky = (float*)ws;

    const int nTiles = (N + 15) / 16;

    gat_gemm_wmma<<<(nTiles + 7) / 8, 256, 0, stream>>>(h, W, Wh, N, nTiles);
    gat_scores<<<(N + 7) / 8, 256, 0, stream>>>(Wh, a, s, t, N);
    gat_init<<<((size_t)N * OUT_F + 255) / 256, 256, 0, stream>>>(seg_max, seg_sum, acc, N);
    gat_edge_max<<<(nE + 255) / 256, 256, 0, stream>>>(ei, s, t, e_leaky, seg_max, nE);
    gat_edge_agg<<<(nE + 7) / 8, 256, 0, stream>>>(ei, e_leaky, seg_max, Wh, seg_sum, acc, nE);
    gat_finalize<<<((size_t)N * OUT_F + 255) / 256, 256, 0, stream>>>(acc, seg_sum, out, N);
}